// CTroidDO_55843164783482
// MI455X (gfx1250) — compile-verified
//
#include <hip/hip_runtime.h>

typedef float v2f __attribute__((ext_vector_type(2)));
typedef float v4f __attribute__((ext_vector_type(4)));
typedef float v8f __attribute__((ext_vector_type(8)));
typedef int   v4i __attribute__((ext_vector_type(4)));

#define BATCH   1024
#define INF     512     // in_features (K)
#define OUTF    512     // out_features (N)
#define KC      64      // K chunk staged in LDS
#define LSTRIDE 68      // KC + 4; ≡ 4 (mod 64) -> conflict-free b64 fragment reads

// Detect gfx1250 async global->LDS builtins (device pass only; host pass and
// older toolchains fall back to the load/ds_store staging path).
#if defined(__HIP_DEVICE_COMPILE__) && __has_builtin(__builtin_amdgcn_global_load_async_to_lds_b128)
#define ASYNC_LDS 1
typedef __attribute__((address_space(1))) v4i* g_v4i_p;   // global b128 pointer
typedef __attribute__((address_space(3))) v4i* l_v4i_p;   // LDS b128 pointer

__device__ __forceinline__ void async_copy16(float* lds_dst, const float* gsrc) {
    // builtin signature (from hipcc diagnostic): (v4i AS1*, v4i AS3*, imm, imm)
    __builtin_amdgcn_global_load_async_to_lds_b128(
        (g_v4i_p)(float*)gsrc, (l_v4i_p)lds_dst, /*offset=*/0, /*cpol=*/0);
}
__device__ __forceinline__ void wait_async0() {
#if __has_builtin(__builtin_amdgcn_s_wait_asynccnt)
    __builtin_amdgcn_s_wait_asynccnt(0);
#else
    asm volatile("s_wait_asynccnt 0x0" ::: "memory");
#endif
}
#endif

// ---------------------------------------------------------------------------
// Pass 1: row squared-norms. One wave (32 lanes) per 512-float row.
// ---------------------------------------------------------------------------
__global__ __launch_bounds__(256)
void rownorm_kernel(const float* __restrict__ src, float* __restrict__ dst, int nrows)
{
    const int gwave = (blockIdx.x * 256 + threadIdx.x) >> 5;
    const int lane  = threadIdx.x & 31;
    if (gwave >= nrows) return;

    const float* row = src + (size_t)gwave * INF;
    float s = 0.0f;
    #pragma unroll
    for (int i = 0; i < 4; ++i) {                 // 128 float4 per row / 32 lanes
        v4f v = *(const v4f*)(row + (i * 32 + lane) * 4);
        s += v.x * v.x + v.y * v.y + v.z * v.z + v.w * v.w;
    }
    #pragma unroll
    for (int off = 16; off > 0; off >>= 1)        // wave32 reduction
        s += __shfl_xor(s, off, 32);
    if (lane == 0) dst[gwave] = s;
}

// ---------------------------------------------------------------------------
// Pass 2: out[b,o] = -gamma[o] * ( dn[b] + wn[o] - 2 * dot(D_b, W_o) )
// One wave per 16x16 tile of D*W^T via V_WMMA_F32_16X16X4_F32.
// 8 waves per block share the 16-wide N tile (shared W staging in LDS).
// ---------------------------------------------------------------------------
__global__ __launch_bounds__(256)
void ctroid_wmma_f32_kernel(const float* __restrict__ D,
                            const float* __restrict__ W,
                            const float* __restrict__ gamma,
                            const float* __restrict__ dnorm,   // (1024,)
                            const float* __restrict__ wnorm,   // (512,)
                            float* __restrict__ out)
{
    __shared__ float smW[16 * LSTRIDE];        // W tile: 16 rows (N) x KC
    __shared__ float smD[8][16 * LSTRIDE];     // per-wave D tile: 16 rows (M) x KC

    const int tid  = threadIdx.x;
    const int wave = tid >> 5;
    const int lane = tid & 31;
    const int hi   = lane >> 4;   // 0: lanes 0-15 (K=0,1); 1: lanes 16-31 (K=2,3)
    const int lo   = lane & 15;

    const int nBase = blockIdx.x * 16;                 // N tile (shared by block)
    const int mBase = (blockIdx.y * 8 + wave) * 16;    // M tile (per wave)

    v8f acc = {};                                      // 16x16 f32 C/D fragment

    // W staging map: one 16B transfer per thread per chunk (16 rows x 16 float4)
    const int wRow = tid >> 4;         // 0..15
    const int wCol = (tid & 15) * 4;   // 0..60

    for (int kc = 0; kc < INF; kc += KC) {
#ifdef ASYNC_LDS
        // gfx1250 async DMA: global -> LDS, tracked on ASYNCcnt
        async_copy16(&smW[wRow * LSTRIDE + wCol],
                     W + (nBase + wRow) * INF + kc + wCol);
        #pragma unroll
        for (int s = 0; s < 8; ++s) {
            const int r = s * 2 + hi;
            const int c = lo * 4;
            async_copy16(&smD[wave][r * LSTRIDE + c],
                         D + (mBase + r) * INF + kc + c);
        }
        wait_async0();
#else
        {
            v4f v = *(const v4f*)(W + (nBase + wRow) * INF + kc + wCol);
            *(v4f*)(&smW[wRow * LSTRIDE + wCol]) = v;
        }
        #pragma unroll
        for (int s = 0; s < 8; ++s) {
            const int r = s * 2 + hi;
            const int c = lo * 4;
            v4f v = *(const v4f*)(D + (mBase + r) * INF + kc + c);
            *(v4f*)(&smD[wave][r * LSTRIDE + c]) = v;
        }
        if (kc + KC < INF) {   // hint next chunk toward the caches
            __builtin_prefetch(W + (nBase + wRow) * INF + kc + KC + wCol, 0, 1);
            __builtin_prefetch(D + (mBase + (hi ? 1 : 0)) * INF + kc + KC + lo * 4, 0, 1);
        }
#endif
        __syncthreads();

        // ---- 16 K-steps of V_WMMA_F32_16X16X4_F32, pure wmma+ds mix ----
        #pragma unroll
        for (int kk = 0; kk < KC; kk += 4) {
            // fp32 A 16x4: lanes 0-15 hold K=0,1; lanes 16-31 hold K=2,3 (float2)
            v2f a = *(const v2f*)(&smD[wave][lo * LSTRIDE + kk + 2 * hi]);
            // fp32 B 4x16 mirrors A: per-lane float2 of rows K=2*hi, 2*hi+1
            v2f b = *(const v2f*)(&smW[lo * LSTRIDE + kk + 2 * hi]);
            acc = __builtin_amdgcn_wmma_f32_16x16x4_f32(
                false, a, false, b, (short)0, acc, false, false);
        }
        __syncthreads();
    }

    // Epilogue: fold precomputed norms + gamma. C fragment: VGPR j holds
    // (M = 8*hi + j, N = lo).
    const float wn = wnorm[nBase + lo];
    const float g  = gamma[nBase + lo];

    #pragma unroll
    for (int j = 0; j < 8; ++j) {
        const int   m   = hi * 8 + j;
        const float dnm = dnorm[mBase + m];
        out[(mBase + m) * OUTF + nBase + lo] = g * (2.0f * acc[j] - dnm - wn);
    }
}

extern "C" void kernel_launch(void* const* d_in, const int* in_sizes, int n_in,
                              void* d_out, int out_size, void* d_ws, size_t ws_size,
                              hipStream_t stream) {
    (void)in_sizes; (void)n_in; (void)ws_size; (void)out_size;
    const float* D     = (const float*)d_in[0];   // (1024, 512)
    const float* W     = (const float*)d_in[1];   // (512, 512)
    const float* gamma = (const float*)d_in[2];   // (512,)
    float*       out   = (float*)d_out;           // (1024, 512)

    float* dnorm = (float*)d_ws;                  // 1024 floats
    float* wnorm = dnorm + BATCH;                 // 512 floats (6 KB total)

    // Pass 1: row norms (8 waves / block)
    rownorm_kernel<<<dim3(BATCH / 8), dim3(256), 0, stream>>>(D, dnorm, BATCH);
    rownorm_kernel<<<dim3(OUTF / 8),  dim3(256), 0, stream>>>(W, wnorm, OUTF);

    // Pass 2: WMMA GEMM + fused distance epilogue
    dim3 grid(OUTF / 16, BATCH / 128);  // 32 x 8 blocks
    dim3 block(256);                    // 8 waves (wave32)
    ctroid_wmma_f32_kernel<<<grid, block, 0, stream>>>(D, W, gamma, dnorm, wnorm, out);
}